// Foreground_fusion_66726611910702
// MI455X (gfx1250) — compile-verified
//
#include <hip/hip_runtime.h>

// ---------------------------------------------------------------------------
// Problem constants (from reference)
// ---------------------------------------------------------------------------
#define HH    192
#define WW    640
#define HWPIX (HH * WW)          // 122880 pixels
#define C0    64
#define C1    128
#define C2    256
#define CI    256
#define KSUM  (C0 + C1)          // 192 = reduced K for the 1x1 branch GEMM

#define NPX    32                // pixels per block (2 WMMA N-tiles)
#define BLKS   (HWPIX / NPX)     // 3840 blocks (WW % 32 == 0)
#define WIN2   102               // conv halo window: 3 rows x 34 cols

// Pair-interleaved LDS panels: panel[k/2][pixel][2] so a B fragment
// (k, k+1 at one pixel) is a single aligned ds_load_b64.
// Pair-row strides are == 32 (mod 64) dwords: kh=0 and kh=1 half-waves
// land on disjoint 32-bank halves -> conflict-free.
#define SFPAIR 96                // fuse pair-row stride (>= 2*NPX, ==32 mod 64)
#define CPAIR  224               // conv pair-row stride (>= 2*WIN2, ==32 mod 64)
#define CONV_LDS_BYTES ((CI / 2) * CPAIR * 4)   // 114,688 B dynamic LDS

typedef __attribute__((ext_vector_type(2))) float v2f;
typedef __attribute__((ext_vector_type(8))) float v8f;

// ---------------------------------------------------------------------------
// Kernel 1: initialize winner-index buffers to -1
// ---------------------------------------------------------------------------
__global__ void init_winners(int* __restrict__ w, int n) {
    int i = blockIdx.x * blockDim.x + threadIdx.x;
    if (i < n) w[i] = -1;
}

// ---------------------------------------------------------------------------
// Kernel 2: last-write-wins scatter == max point index wins -> atomicMax
// ---------------------------------------------------------------------------
__global__ void scatter_winners(const int* __restrict__ g0,
                                const int* __restrict__ g1,
                                const int* __restrict__ g2,
                                int* __restrict__ w0,
                                int* __restrict__ w1,
                                int* __restrict__ w2,
                                int npts) {
    int i = blockIdx.x * blockDim.x + threadIdx.x;
    if (i >= npts) return;
    int c, r;
    c = g0[2 * i]; r = g0[2 * i + 1];
    if ((unsigned)c < (unsigned)WW && (unsigned)r < (unsigned)HH)
        atomicMax(&w0[r * WW + c], i);
    c = g1[2 * i]; r = g1[2 * i + 1];
    if ((unsigned)c < (unsigned)WW && (unsigned)r < (unsigned)HH)
        atomicMax(&w1[r * WW + c], i);
    c = g2[2 * i]; r = g2[2 * i + 1];
    if ((unsigned)c < (unsigned)WW && (unsigned)r < (unsigned)HH)
        atomicMax(&w2[r * WW + c], i);
}

// ---------------------------------------------------------------------------
// Kernel 2b: one-time weight repack for the 3x3 conv:
//   Wsb[o][k][tap]  ->  Wt[tap][o][k]
// A-fragment loads become a single aligned global_load_b64 per WMMA step.
// ---------------------------------------------------------------------------
__global__ void repack_wsb(const float* __restrict__ Wsb,
                           float* __restrict__ Wt) {
    int i = blockIdx.x * blockDim.x + threadIdx.x;   // over 9*256*256
    if (i >= 9 * CI * CI) return;
    const int tap = i / (CI * CI);
    const int rem = i % (CI * CI);
    const int o   = rem / CI;
    const int k   = rem % CI;
    Wt[i] = Wsb[(o * CI + k) * 9 + tap];
}

// ---------------------------------------------------------------------------
// Kernel 3: gather point feats + 1x1 convs + branch sum  (WMMA f32 16x16x4)
// Block: 512 threads = 16 waves; wave w -> out-channels [16w, 16w+16).
// Block covers 32 pixels = 2 N-tiles: each A fragment feeds 2 WMMAs
// (halved weight traffic) on 2 independent accumulator chains.
// Inner loop per k-step: 1 global_load_b64 (A) + 2 ds_load_b64 (B) + 2 WMMA.
// ---------------------------------------------------------------------------
__global__ __launch_bounds__(512)
void fuse_pointwise(const float* __restrict__ vf0,
                    const float* __restrict__ vf1,
                    const float* __restrict__ vf2,
                    const int*   __restrict__ w0,
                    const int*   __restrict__ w1,
                    const int*   __restrict__ w2,
                    const float* __restrict__ Wrd0,
                    const float* __restrict__ brd0,
                    const float* __restrict__ Wrd1,
                    const float* __restrict__ brd1,
                    float* __restrict__ pt_img) {
    __shared__ float sf[(KSUM / 2) * SFPAIR];  // pair-interleaved feat panel
    __shared__ int   sw0[NPX], sw1[NPX], sw2[NPX];

    const int p0   = blockIdx.x * NPX;      // base pixel of this block
    const int t    = threadIdx.x;
    const int wave = t >> 5;
    const int lane = t & 31;
    const int ml   = lane & 15;             // matrix row/col within half-wave
    const int kh   = lane >> 4;             // 0: K pair {0,1}, 1: K pair {2,3}
    const int m0   = wave * 16;             // out-channel base of this wave

    if (t < NPX) {
        sw0[t] = w0[p0 + t];
        sw1[t] = w1[p0 + t];
        sw2[t] = w2[p0 + t];
    }
    __syncthreads();

    // stage gathered features as (k, k+1) pairs: 96 pairs x 32 pixels
    for (int e = t; e < (KSUM / 2) * NPX; e += 512) {
        const int pair = e >> 5;
        const int p    = e & 31;
        const int k    = pair * 2;
        v2f pr; pr.x = 0.0f; pr.y = 0.0f;
        if (k < C0) {
            const int w = sw0[p];
            if (w >= 0) pr = *(const v2f*)(vf0 + w * C0 + k);
        } else {
            const int w = sw1[p];
            if (w >= 0) pr = *(const v2f*)(vf1 + w * C1 + (k - C0));
        }
        *(v2f*)&sf[pair * SFPAIR + 2 * p] = pr;
    }
    __syncthreads();

    const float* wa0 = Wrd0 + (m0 + ml) * C0 + kh * 2;
    const float* wa1 = Wrd1 + (m0 + ml) * C1 + kh * 2;
    const float* sfb = sf + kh * SFPAIR + 2 * ml;   // pair index += kh

    v8f acc0 = {}, acc1 = {};
#pragma unroll 4
    for (int kk = 0; kk < C0; kk += 4) {
        const v2f a  = *(const v2f*)(wa0 + kk);                    // A: b64
        const v2f b0 = *(const v2f*)(sfb + (kk >> 1) * SFPAIR);    // B: b64
        const v2f b1 = *(const v2f*)(sfb + (kk >> 1) * SFPAIR + 32);
        acc0 = __builtin_amdgcn_wmma_f32_16x16x4_f32(
                   false, a, false, b0, (short)0, acc0, false, false);
        acc1 = __builtin_amdgcn_wmma_f32_16x16x4_f32(
                   false, a, false, b1, (short)0, acc1, false, false);
    }
#pragma unroll 4
    for (int kk = 0; kk < C1; kk += 4) {
        const v2f a  = *(const v2f*)(wa1 + kk);
        const v2f b0 = *(const v2f*)(sfb + ((C0 + kk) >> 1) * SFPAIR);
        const v2f b1 = *(const v2f*)(sfb + ((C0 + kk) >> 1) * SFPAIR + 32);
        acc0 = __builtin_amdgcn_wmma_f32_16x16x4_f32(
                   false, a, false, b0, (short)0, acc0, false, false);
        acc1 = __builtin_amdgcn_wmma_f32_16x16x4_f32(
                   false, a, false, b1, (short)0, acc1, false, false);
    }

    // epilogue: biases + raw 256-ch branch gather, store both D fragments
    const int px0 = p0 + ml;
    const int px1 = p0 + 16 + ml;
    const int wv0 = sw2[ml];
    const int wv1 = sw2[ml + 16];
#pragma unroll
    for (int r = 0; r < 8; ++r) {
        const int c = m0 + r + kh * 8;      // D: M = r + 8*(lane>=16)
        const float bias = brd0[c] + brd1[c];
        float v0 = acc0[r] + bias;
        float v1 = acc1[r] + bias;
        if (wv0 >= 0) v0 += vf2[wv0 * C2 + c];
        if (wv1 >= 0) v1 += vf2[wv1 * C2 + c];
        pt_img[c * HWPIX + px0] = v0;
        pt_img[c * HWPIX + px1] = v1;
    }
}

// ---------------------------------------------------------------------------
// Kernel 4: 3x3 conv 256->256 (implicit GEMM, WMMA f32) fused with
//   masked_feat = (conv + b_sb + img_feat) * seg_mask
// Block: 512 threads = 16 waves (all 256 out channels), 32-pixel strip.
// Zero-padded 3x34 halo window staged pair-interleaved in ~112 KB dynamic
// LDS; the 9-tap x 64-kblock loop has no bounds checks and no register
// shuffles: per k-step 1 global_load_b64 (A, L2-hot repacked weights) +
// 2 ds_load_b64 (B) + 2 WMMAs on independent accumulator chains.
// ---------------------------------------------------------------------------
__global__ __launch_bounds__(512)
void conv3x3_wmma(const float* __restrict__ pt_img,
                  const float* __restrict__ Wt,    // repacked [9][256][256]
                  const float* __restrict__ bsb,
                  const float* __restrict__ img_feat,
                  const float* __restrict__ seg_prob,  // [2][H][W], use ch 1
                  float* __restrict__ masked_feat) {
    extern __shared__ float sb[];     // [pair][window pixel][2]

    const int tile = blockIdx.x;
    const int ty   = tile / (WW / NPX);
    const int tx   = (tile % (WW / NPX)) * NPX;
    const int t    = threadIdx.x;
    const int wave = t >> 5;
    const int lane = t & 31;
    const int ml   = lane & 15;
    const int kh   = lane >> 4;
    const int m0   = wave * 16;

    // stage zero-padded halo window as channel pairs: 128 pairs x 102 pixels
    for (int e = t; e < (CI / 2) * WIN2; e += 512) {
        const int pair = e / WIN2;
        const int wp   = e % WIN2;
        const int k    = pair * 2;
        const int wrow = wp / 34;           // 0..2
        const int wcol = wp % 34;           // 0..33
        const int y    = ty + wrow - 1;
        const int x    = tx + wcol - 1;
        v2f pr; pr.x = 0.0f; pr.y = 0.0f;
        if (y >= 0 && y < HH && x >= 0 && x < WW) {
            pr.x = pt_img[(k + 0) * HWPIX + y * WW + x];
            pr.y = pt_img[(k + 1) * HWPIX + y * WW + x];
        }
        *(v2f*)&sb[pair * CPAIR + 2 * wp] = pr;
    }
    __syncthreads();

    v8f acc0 = {}, acc1 = {};
#pragma unroll
    for (int tap = 0; tap < 9; ++tap) {
        const int dyi = tap / 3;            // 0..2
        const int dxi = tap % 3;            // 0..2
        const float* wt  = Wt + ((tap * CI) + (m0 + ml)) * CI + kh * 2;
        const float* sbp = sb + kh * CPAIR + 2 * (dyi * 34 + dxi + ml);
#pragma unroll 4
        for (int kk = 0; kk < CI; kk += 4) {
            const v2f a  = *(const v2f*)(wt + kk);                  // A: b64
            const v2f b0 = *(const v2f*)(sbp + (kk >> 1) * CPAIR);  // B: b64
            const v2f b1 = *(const v2f*)(sbp + (kk >> 1) * CPAIR + 32);
            acc0 = __builtin_amdgcn_wmma_f32_16x16x4_f32(
                       false, a, false, b0, (short)0, acc0, false, false);
            acc1 = __builtin_amdgcn_wmma_f32_16x16x4_f32(
                       false, a, false, b1, (short)0, acc1, false, false);
        }
    }

    const int pix0 = ty * WW + tx + ml;
    const int pix1 = pix0 + 16;
    const float mask0 = (seg_prob[HWPIX + pix0] > 0.5f) ? 1.0f : 0.0f;
    const float mask1 = (seg_prob[HWPIX + pix1] > 0.5f) ? 1.0f : 0.0f;
#pragma unroll
    for (int r = 0; r < 8; ++r) {
        const int c = m0 + r + kh * 8;
        const float bias = bsb[c];
        masked_feat[c * HWPIX + pix0] =
            (acc0[r] + bias + img_feat[c * HWPIX + pix0]) * mask0;
        masked_feat[c * HWPIX + pix1] =
            (acc1[r] + bias + img_feat[c * HWPIX + pix1]) * mask1;
    }
}

// ---------------------------------------------------------------------------
// Kernel 5: attention conv (256->1, 3x3) + sigmoid + broadcast multiply
// ---------------------------------------------------------------------------
__global__ __launch_bounds__(256)
void att_out(const float* __restrict__ masked_feat,
             const float* __restrict__ Wcv,   // [1][256][3][3]
             const float* __restrict__ bcv,
             const float* __restrict__ img_feat,
             float* __restrict__ out) {
    const int pix = blockIdx.x * blockDim.x + threadIdx.x;
    if (pix >= HWPIX) return;
    const int y = pix / WW;
    const int x = pix % WW;

    // hoisted border predicates / offsets (loop-invariant in c)
    bool valid[9];
    int  off[9];
#pragma unroll
    for (int tap = 0; tap < 9; ++tap) {
        const int py = y + tap / 3 - 1;
        const int px = x + tap % 3 - 1;
        valid[tap] = (py >= 0 && py < HH && px >= 0 && px < WW);
        off[tap]   = valid[tap] ? (py * WW + px) : 0;
    }

    float s = bcv[0];
    for (int c = 0; c < CI; ++c) {
        const float* mf = masked_feat + c * HWPIX;
        const float* wk = Wcv + c * 9;
#pragma unroll
        for (int tap = 0; tap < 9; ++tap) {
            const float v = valid[tap] ? mf[off[tap]] : 0.0f;
            s += wk[tap] * v;
        }
    }
    const float att = 1.0f / (1.0f + __expf(-s));
    for (int c = 0; c < CI; ++c)
        out[c * HWPIX + pix] = att * img_feat[c * HWPIX + pix];
}

// ---------------------------------------------------------------------------
// Host-side launch
// ---------------------------------------------------------------------------
extern "C" void kernel_launch(void* const* d_in, const int* in_sizes, int n_in,
                              void* d_out, int out_size, void* d_ws, size_t ws_size,
                              hipStream_t stream) {
    const float* img_feat = (const float*)d_in[0];
    const float* seg_prob = (const float*)d_in[1];
    const float* vf0      = (const float*)d_in[2];
    const int*   g0       = (const int*)  d_in[3];
    const float* vf1      = (const float*)d_in[4];
    const int*   g1       = (const int*)  d_in[5];
    const float* vf2      = (const float*)d_in[6];
    const int*   g2       = (const int*)  d_in[7];
    const float* Wrd0     = (const float*)d_in[8];
    const float* brd0     = (const float*)d_in[9];
    const float* Wrd1     = (const float*)d_in[10];
    const float* brd1     = (const float*)d_in[11];
    const float* Wsb      = (const float*)d_in[12];
    const float* bsb      = (const float*)d_in[13];
    const float* Wcv      = (const float*)d_in[14];
    const float* bcv      = (const float*)d_in[15];
    float* out = (float*)d_out;

    const int npts = in_sizes[3] / 2;

    // workspace layout
    char*  ws     = (char*)d_ws;
    int*   w0     = (int*)ws;
    int*   w1     = w0 + HWPIX;
    int*   w2     = w1 + HWPIX;
    float* pt_img = (float*)(ws + (size_t)3 * HWPIX * sizeof(int));
    float* masked = pt_img + (size_t)CI * HWPIX;
    float* Wt     = masked + (size_t)CI * HWPIX;    // repacked conv weights

    init_winners<<<(3 * HWPIX + 255) / 256, 256, 0, stream>>>(w0, 3 * HWPIX);
    scatter_winners<<<(npts + 255) / 256, 256, 0, stream>>>(g0, g1, g2,
                                                            w0, w1, w2, npts);
    repack_wsb<<<(9 * CI * CI + 255) / 256, 256, 0, stream>>>(Wsb, Wt);
    fuse_pointwise<<<BLKS, 512, 0, stream>>>(vf0, vf1, vf2, w0, w1, w2,
                                             Wrd0, brd0, Wrd1, brd1, pt_img);
    conv3x3_wmma<<<BLKS, 512, CONV_LDS_BYTES, stream>>>(pt_img, Wt, bsb,
                                                        img_feat, seg_prob,
                                                        masked);
    att_out<<<(HWPIX + 255) / 256, 256, 0, stream>>>(masked, Wcv, bcv,
                                                     img_feat, out);
}